// SubGraphMapping_82145544503996
// MI455X (gfx1250) — compile-verified
//
#include <hip/hip_runtime.h>
#include <math.h>

// ---------------------------------------------------------------------------
// Shapes: B=16384, SEQ=41, K_GROUPS=8, M=5, SITE=20
// d_out layout (float, concatenated in return order):
//   out1 x_bag   [B,8,5,41]  = B*1640   @ 0
//   out2 Asite_x [B,1,41]    = B*41     @ B*1640
//   out3 x_site  [B,41,41]   = B*1681   @ B*1681
//   out4 index   [B,8,5]     = B*40     @ B*3362
// ---------------------------------------------------------------------------

typedef float v2f __attribute__((ext_vector_type(2)));
typedef float v8f __attribute__((ext_vector_type(8)));

#define SEQ     41
#define SITE    20
#define KPAD    124      // 3*41 = 123 padded to multiple of 4
#define WPITCH  48       // LDS pitch for W  [kk][o]
#define XPITCH  52       // LDS pitch for Xt [i][1+h]  (cols 0..49 touched)
#define XROWS   42       // 41 real rows + zero row 41 (absorbs kk==123 pad)
#define YPITCH  44       // LDS pitch for Y  [o][h]
#define WAVES   4        // batches (waves) per block in the WMMA kernel

// ---------------------------------------------------------------------------
// Kernel 1: subgraph selection (1 thread per batch, serial top-k with 41-bit
// mask; matches jax.lax.top_k tie-breaking: strict '>' keeps lowest index).
// ---------------------------------------------------------------------------
__global__ void select_kernel(const float* __restrict__ att,
                              const float* __restrict__ deg,
                              int* __restrict__ idx_ws,
                              float* __restrict__ out4, int B) {
  int b = blockIdx.x * blockDim.x + threadIdx.x;
  if (b >= B) return;

  float d[SEQ];
  for (int j = 0; j < SEQ; ++j) d[j] = deg[(size_t)b * SEQ + j];

  int bsel[8];
  unsigned long long used = 0ull;
  for (int i = 0; i < 8; ++i) {
    float best = -INFINITY; int bi = 0;
    for (int j = 0; j < SEQ; ++j) {
      if (!((used >> j) & 1ull) && d[j] > best) { best = d[j]; bi = j; }
    }
    used |= 1ull << bi;
    bsel[i] = bi;
  }

  unsigned long long sel = 1ull << SITE;   // site 20 pre-selected
  for (int g = 0; g < 8; ++g) {
    float p[SEQ];
    const float* row = att + (size_t)b * (SEQ * SEQ) + (size_t)bsel[g] * SEQ;
    for (int j = 0; j < SEQ; ++j) p[j] = row[j];   // /5.0 is monotonic: skip
    int picked[5];
    for (int m = 0; m < 5; ++m) {
      float best = -INFINITY; int bi = 0;
      for (int j = 0; j < SEQ; ++j) {
        if (!((sel >> j) & 1ull) && p[j] > best) { best = p[j]; bi = j; }
      }
      sel |= 1ull << bi;
      picked[m] = bi;
    }
    // sort 5 indices ascending (jnp.sort within group)
    for (int a = 0; a < 5; ++a)
      for (int c = a + 1; c < 5; ++c)
        if (picked[c] < picked[a]) { int t = picked[a]; picked[a] = picked[c]; picked[c] = t; }
    for (int m = 0; m < 5; ++m) {
      idx_ws[(size_t)b * 40 + g * 5 + m] = picked[m];
      out4[(size_t)b * 40 + g * 5 + m] = (float)picked[m];
    }
  }
}

// ---------------------------------------------------------------------------
// Kernel 2: gather context rows + conv2d 8->8ch 3x3 over [5,41], LDS staged.
// One block per batch.
// ---------------------------------------------------------------------------
__global__ void __launch_bounds__(256)
conv2d_kernel(const float* __restrict__ x, const float* __restrict__ w2d,
              const float* __restrict__ b2d, const int* __restrict__ idx_ws,
              float* __restrict__ out1, int B) {
  __shared__ float sc[8 * 7 * 44];   // padded context [c][1+h][1+w]
  __shared__ float sw[8 * 8 * 9];
  __shared__ float sb[8];
  __shared__ int   sidx[40];

  const int b = blockIdx.x;
  const int tid = threadIdx.x;

  for (int t = tid; t < 8 * 7 * 44; t += 256) sc[t] = 0.0f;
  for (int t = tid; t < 576; t += 256) sw[t] = w2d[t];
  if (tid < 8)  sb[tid] = b2d[tid];
  if (tid < 40) sidx[tid] = idx_ws[(size_t)b * 40 + tid];
  __syncthreads();

  for (int t = tid; t < 8 * 5 * SEQ; t += 256) {
    int w = t % SEQ; int hh = (t / SEQ) % 5; int c = t / (5 * SEQ);
    int row = sidx[c * 5 + hh];
    sc[(c * 7 + hh + 1) * 44 + 1 + w] = x[(size_t)b * (SEQ * SEQ) + (size_t)row * SEQ + w];
  }
  __syncthreads();

  for (int t = tid; t < 8 * 5 * SEQ; t += 256) {
    int w = t % SEQ; int h = (t / SEQ) % 5; int oc = t / (5 * SEQ);
    float acc = sb[oc];
#pragma unroll
    for (int c = 0; c < 8; ++c)
#pragma unroll
      for (int kh = 0; kh < 3; ++kh)
#pragma unroll
        for (int kw = 0; kw < 3; ++kw)
          acc += sc[(c * 7 + h + kh) * 44 + w + kw] * sw[((oc * 8 + c) * 3 + kh) * 3 + kw];
    out1[(size_t)b * 1640 + t] = acc;
  }
}

// ---------------------------------------------------------------------------
// Kernel 3: Asite_x row copy (streaming).
// ---------------------------------------------------------------------------
__global__ void asite_kernel(const float* __restrict__ x, float* __restrict__ out2, int n) {
  int e = blockIdx.x * blockDim.x + threadIdx.x;
  if (e >= n) return;
  int b = e / SEQ, j = e - b * SEQ;
  out2[e] = x[(size_t)b * (SEQ * SEQ) + SITE * SEQ + j];
}

// ---------------------------------------------------------------------------
// Kernel 4: conv1d (41ch->41ch, k=3) + tanh(tanh(.)) as fp32 WMMA GEMM.
//   Y[h,o] = sum_kk A[h,kk] * W[kk,o],  kk = k*41+i (K=123 padded to 124)
//   A[h,kk] = x[b, h+k-1, i]  (zero halo),  W[kk,o] = w1d[o,i,k]
// Branch-free K padding: kk==123 decomposes to (k=2,i=41) and lands in the
// all-zero Xt row 41; W row 123 is zero-filled. One wave per batch,
// 4 waves per block, 3x3 tiles of V_WMMA_F32_16X16X4_F32.
// ---------------------------------------------------------------------------
__global__ void __launch_bounds__(32 * WAVES)
conv1d_wmma_kernel(const float* __restrict__ x, const float* __restrict__ w1d,
                   const float* __restrict__ b1d, float* __restrict__ out3, int B) {
  __shared__ float sW[KPAD * WPITCH];              // [kk][o]
  __shared__ float sX[WAVES][XROWS * XPITCH];      // per-wave Xt[i][1+h]

  const int tid  = threadIdx.x;
  const int wave = tid >> 5;
  const int lane = tid & 31;
  const int half = lane >> 4;    // 0: K={0,1}, 1: K={2,3}
  const int lh   = lane & 15;
  const int b    = blockIdx.x * WAVES + wave;

  // Stage W in a single full-coverage pass (pads -> 0), no extra barrier.
  for (int t = tid; t < KPAD * WPITCH; t += 32 * WAVES) {
    const int kk = t / WPITCH, o = t - kk * WPITCH;
    float v = 0.0f;
    if (kk < SEQ * 3 && o < SEQ) {
      const int k = kk / SEQ, i = kk - k * SEQ;
      v = w1d[(o * SEQ + i) * 3 + k];              // w1d[o][i][k]
    }
    sW[t] = v;
  }

  float* xt = &sX[wave][0];
  if (b < B) {
    // Xt[i][1+h] = x[b][h][i] (coalesced global read, strided LDS write).
    for (int t = lane; t < SEQ * SEQ; t += 32) {
      const int h = t / SEQ, i = t - h * SEQ;
      xt[i * XPITCH + 1 + h] = x[(size_t)b * (SEQ * SEQ) + t];
    }
    // Halo columns 0 and 42 of rows 0..40, and the whole zero row 41.
    for (int i = lane; i < SEQ; i += 32) {
      xt[i * XPITCH + 0]  = 0.0f;
      xt[i * XPITCH + 42] = 0.0f;
    }
    for (int c = lane; c < XPITCH; c += 32) xt[SEQ * XPITCH + c] = 0.0f;
  }
  __syncthreads();
  if (b >= B) return;

  v8f acc[3][3];
#pragma unroll
  for (int mt = 0; mt < 3; ++mt)
#pragma unroll
    for (int nt = 0; nt < 3; ++nt)
#pragma unroll
      for (int e = 0; e < 8; ++e) acc[mt][nt][e] = 0.0f;

  for (int kk0 = 0; kk0 < KPAD; kk0 += 4) {
    const int kkA = kk0 + 2 * half;   // this lane's first K index
    const int kk1 = kkA + 1;
    const int kA0 = (kkA >= 82) ? 2 : ((kkA >= 41) ? 1 : 0);
    const int iA0 = kkA - 41 * kA0;                 // kk==123 -> i=41 (zeros)
    const int kA1 = (kk1 >= 82) ? 2 : ((kk1 >= 41) ? 1 : 0);
    const int iA1 = kk1 - 41 * kA1;
    const int a0  = iA0 * XPITCH + kA0;
    const int a1  = iA1 * XPITCH + kA1;

    v2f afrag[3], bfrag[3];
#pragma unroll
    for (int mt = 0; mt < 3; ++mt) {
      const int h = mt * 16 + lh;                   // A row (M)
      afrag[mt].x = xt[a0 + h];
      afrag[mt].y = xt[a1 + h];
    }
#pragma unroll
    for (int nt = 0; nt < 3; ++nt) {
      const int o = nt * 16 + lh;                   // B col (N)
      bfrag[nt].x = sW[kkA * WPITCH + o];           // row 123 is zero
      bfrag[nt].y = sW[kk1 * WPITCH + o];
    }
#pragma unroll
    for (int mt = 0; mt < 3; ++mt)
#pragma unroll
      for (int nt = 0; nt < 3; ++nt)
        acc[mt][nt] = __builtin_amdgcn_wmma_f32_16x16x4_f32(
            false, afrag[mt], false, bfrag[nt],
            (short)0, acc[mt][nt], false, false);
  }

  // Epilogue: bias + tanh(tanh(.)), transpose through LDS (reuse xt), store.
  float bias[3];
#pragma unroll
  for (int nt = 0; nt < 3; ++nt) {
    int o = nt * 16 + lh;
    bias[nt] = (o < SEQ) ? b1d[o] : 0.0f;
  }
  float* yl = xt;   // [o][h] with pitch YPITCH; all xt reads are done above
#pragma unroll
  for (int mt = 0; mt < 3; ++mt) {
#pragma unroll
    for (int nt = 0; nt < 3; ++nt) {
      const int o = nt * 16 + lh;
#pragma unroll
      for (int r = 0; r < 8; ++r) {
        const int h = mt * 16 + 8 * half + r;
        if (h < SEQ && o < SEQ) {
          float v = acc[mt][nt][r] + bias[nt];
          v = tanhf(tanhf(v));
          yl[o * YPITCH + h] = v;
        }
      }
    }
  }
  for (int t = lane; t < SEQ * SEQ; t += 32) {
    int o = t / SEQ, h = t - o * SEQ;
    out3[(size_t)b * (SEQ * SEQ) + t] = yl[o * YPITCH + h];
  }
}

// ---------------------------------------------------------------------------
extern "C" void kernel_launch(void* const* d_in, const int* in_sizes, int n_in,
                              void* d_out, int out_size, void* d_ws, size_t ws_size,
                              hipStream_t stream) {
  const float* att = (const float*)d_in[0];   // [B,41,41]
  const float* deg = (const float*)d_in[1];   // [B,41]
  const float* x   = (const float*)d_in[2];   // [B,41,41]
  const float* w1d = (const float*)d_in[3];   // [41,41,3]
  const float* b1d = (const float*)d_in[4];   // [41]
  const float* w2d = (const float*)d_in[5];   // [8,8,3,3]
  const float* b2d = (const float*)d_in[6];   // [8]

  const int B = in_sizes[1] / SEQ;

  float* out  = (float*)d_out;
  float* out1 = out;                           // x_bag   B*1640
  float* out2 = out + (size_t)B * 1640;        // Asite_x B*41
  float* out3 = out + (size_t)B * 1681;        // x_site  B*1681
  float* out4 = out + (size_t)B * 3362;        // index   B*40
  int*   idx_ws = (int*)d_ws;                  // B*40 int32

  select_kernel<<<(B + 255) / 256, 256, 0, stream>>>(att, deg, idx_ws, out4, B);
  conv2d_kernel<<<B, 256, 0, stream>>>(x, w2d, b2d, idx_ws, out1, B);
  const int n2 = B * SEQ;
  asite_kernel<<<(n2 + 255) / 256, 256, 0, stream>>>(x, out2, n2);
  conv1d_wmma_kernel<<<(B + WAVES - 1) / WAVES, 32 * WAVES, 0, stream>>>(x, w1d, b1d, out3, B);
}